// JK_SAGE_GDQN_Attention_72851235275290
// MI455X (gfx1250) — compile-verified
//
#include <hip/hip_runtime.h>

typedef __attribute__((ext_vector_type(2))) float v2f;
typedef __attribute__((ext_vector_type(8))) float v8f;

static constexpr int  BGRAPH = 16384;
static constexpr int  NNODES = 55;
static constexpr long NTOT   = (long)BGRAPH * NNODES;   // 901120
static constexpr long EDGES  = 4 * NTOT;                // 3604480
static constexpr int  HID    = 32;
static constexpr int  INCH   = 13;

__device__ __forceinline__ float selu_f(float x) {
    const float alpha = 1.6732632423543772f;
    const float scale = 1.0507009873554805f;
    return scale * (x > 0.f ? x : alpha * (__expf(x) - 1.f));
}

__device__ __forceinline__ v8f wmma_f32(v2f a, v2f b, v8f c) {
    // V_WMMA_F32_16X16X4_F32: D = A(16x4) * B(4x16) + C(16x16), fp32
    return __builtin_amdgcn_wmma_f32_16x16x4_f32(
        /*neg_a=*/false, a, /*neg_b=*/false, b,
        /*c_mod=*/(short)0, c, /*reuse_a=*/false, /*reuse_b=*/false);
}

// ---------------------------------------------------------------- utilities
__global__ void zero_f4(float4* __restrict__ p, long n4) {
    long i = (long)blockIdx.x * blockDim.x + threadIdx.x;
    if (i < n4) p[i] = make_float4(0.f, 0.f, 0.f, 0.f);
}

__global__ void selu_inplace(float* __restrict__ p, long n) {
    long i = (long)blockIdx.x * blockDim.x + threadIdx.x;
    if (i < n) p[i] = selu_f(p[i]);
}

// in-degree (float, matches reference cnt semantics)
__global__ void count_deg(const int* __restrict__ dst, float* __restrict__ cnt, long e) {
    long i = (long)blockIdx.x * blockDim.x + threadIdx.x;
    if (i < e) atomicAdd(&cnt[dst[i]], 1.0f);
}

// layer-0 scatter: x rows (stride 13) -> agg rows (stride 16, cols 13..15 stay 0)
__global__ void scatter13(const float* __restrict__ x, const int* __restrict__ src,
                          const int* __restrict__ dst, float* __restrict__ agg, long e) {
    long idx = (long)blockIdx.x * blockDim.x + threadIdx.x;
    if (idx >= e * 16) return;
    long eid = idx >> 4;
    int  c   = (int)(idx & 15);
    if (c < INCH) {
        long s = src[eid], d = dst[eid];
        atomicAdd(&agg[d * 16 + c], x[s * INCH + c]);
    }
}

// hidden scatter: h rows (stride 32) -> agg rows (stride 32); a wave covers one full row
__global__ void scatter32(const float* __restrict__ h, const int* __restrict__ src,
                          const int* __restrict__ dst, float* __restrict__ agg, long e) {
    long idx = (long)blockIdx.x * blockDim.x + threadIdx.x;
    if (idx >= e * 32) return;
    long eid = idx >> 5;
    int  c   = (int)(idx & 31);
    long s = src[eid], d = dst[eid];
    atomicAdd(&agg[d * 32 + c], h[s * 32 + c]);
}

// ------------------------------------------------- WMMA SAGE node update
// out = selu( (agg/max(cnt,1)) @ Wl + bl + prev @ Wr );  jk = FIRST ? out : max(jk,out)
// One wave = 16 nodes x 32 cols. 8 waves/block = 128 nodes/block. N % 128 == 0.
template<int KSTEPS, int KD, int PS, int AS, bool FIRST>
__global__ __launch_bounds__(256)
void sage_node(const float* __restrict__ agg, const float* __restrict__ cnt,
               const float* __restrict__ prev,
               const float* __restrict__ wl, const float* __restrict__ bl,
               const float* __restrict__ wr,
               float* __restrict__ hout, float* __restrict__ jk) {
    __shared__ float wlS[KSTEPS * 4 * HID];
    __shared__ float wrS[KSTEPS * 4 * HID];
    __shared__ float bS[HID];
    const int tid = threadIdx.x;
    for (int i = tid; i < KSTEPS * 4 * HID; i += 256) {
        wlS[i] = (i < KD * HID) ? wl[i] : 0.f;
        wrS[i] = (i < KD * HID) ? wr[i] : 0.f;
    }
    if (tid < HID) bS[tid] = bl[tid];
    __syncthreads();

    const int wave = tid >> 5, lane = tid & 31;
    const int mlo = lane & 15, hi = lane >> 4;
    const long nodeBase = (long)blockIdx.x * 128 + wave * 16;
    const long row = nodeBase + mlo;
    const float* __restrict__ arow = agg + row * AS;
    const float* __restrict__ prow = prev + row * PS;
    const float inv = 1.0f / fmaxf(cnt[row], 1.0f);

    v8f c0 = {};   // cols 0..15
    v8f c1 = {};   // cols 16..31
#pragma unroll
    for (int kk = 0; kk < KSTEPS; ++kk) {
        const int kb = kk * 4 + hi * 2;   // A/B fragment K offsets (lanes 0-15: +0,+1; 16-31: +2,+3)
        v2f a;
        a.x = arow[kb] * inv;             // AS-padded region is zeroed
        a.y = arow[kb + 1] * inv;
        v2f b0, b1;
        b0.x = wlS[kb * HID + mlo];        b0.y = wlS[(kb + 1) * HID + mlo];
        b1.x = wlS[kb * HID + 16 + mlo];   b1.y = wlS[(kb + 1) * HID + 16 + mlo];
        c0 = wmma_f32(a, b0, c0);
        c1 = wmma_f32(a, b1, c1);

        v2f a2;
        a2.x = (kb     < KD) ? prow[kb]     : 0.f;   // prev is packed (stride 13 for layer0)
        a2.y = (kb + 1 < KD) ? prow[kb + 1] : 0.f;
        v2f r0, r1;
        r0.x = wrS[kb * HID + mlo];        r0.y = wrS[(kb + 1) * HID + mlo];
        r1.x = wrS[kb * HID + 16 + mlo];   r1.y = wrS[(kb + 1) * HID + 16 + mlo];
        c0 = wmma_f32(a2, r0, c0);
        c1 = wmma_f32(a2, r1, c1);
    }
    // C layout: VGPR j -> row j (lanes 0-15) / row j+8 (lanes 16-31), col = (lane&15)+16*half
#pragma unroll
    for (int j = 0; j < 8; ++j) {
        const long rj = nodeBase + j + 8 * hi;
        const float v0 = selu_f(c0[j] + bS[mlo]);
        const float v1 = selu_f(c1[j] + bS[16 + mlo]);
        const long o0 = rj * HID + mlo, o1 = rj * HID + 16 + mlo;
        hout[o0] = v0; hout[o1] = v1;
        if (FIRST) { jk[o0] = v0; jk[o1] = v1; }
        else       { jk[o0] = fmaxf(jk[o0], v0); jk[o1] = fmaxf(jk[o1], v1); }
    }
}

// ------------------------------------------------- fused K/V projection (WMMA)
__global__ __launch_bounds__(256)
void proj_kv(const float* __restrict__ keys,
             const float* __restrict__ wk, const float* __restrict__ bk,
             const float* __restrict__ wv, const float* __restrict__ bv,
             float* __restrict__ Kp, float* __restrict__ Vp) {
    __shared__ float wkS[HID * HID];
    __shared__ float wvS[HID * HID];
    __shared__ float bkS[HID], bvS[HID];
    const int tid = threadIdx.x;
    for (int i = tid; i < HID * HID; i += 256) { wkS[i] = wk[i]; wvS[i] = wv[i]; }
    if (tid < HID) { bkS[tid] = bk[tid]; bvS[tid] = bv[tid]; }
    __syncthreads();

    const int wave = tid >> 5, lane = tid & 31;
    const int mlo = lane & 15, hi = lane >> 4;
    const long nodeBase = (long)blockIdx.x * 128 + wave * 16;
    const float* __restrict__ arow = keys + (nodeBase + mlo) * HID;

    v8f k0 = {}, k1 = {}, v0a = {}, v1a = {};
#pragma unroll
    for (int kk = 0; kk < 8; ++kk) {
        const int kb = kk * 4 + hi * 2;
        v2f a;  a.x = arow[kb];  a.y = arow[kb + 1];
        v2f b0, b1, c0, c1;
        b0.x = wkS[kb * HID + mlo];        b0.y = wkS[(kb + 1) * HID + mlo];
        b1.x = wkS[kb * HID + 16 + mlo];   b1.y = wkS[(kb + 1) * HID + 16 + mlo];
        c0.x = wvS[kb * HID + mlo];        c0.y = wvS[(kb + 1) * HID + mlo];
        c1.x = wvS[kb * HID + 16 + mlo];   c1.y = wvS[(kb + 1) * HID + 16 + mlo];
        k0  = wmma_f32(a, b0, k0);
        k1  = wmma_f32(a, b1, k1);
        v0a = wmma_f32(a, c0, v0a);
        v1a = wmma_f32(a, c1, v1a);
    }
#pragma unroll
    for (int j = 0; j < 8; ++j) {
        const long rj = nodeBase + j + 8 * hi;
        const long o0 = rj * HID + mlo, o1 = rj * HID + 16 + mlo;
        Kp[o0] = k0[j]  + bkS[mlo];       Kp[o1] = k1[j]  + bkS[16 + mlo];
        Vp[o0] = v0a[j] + bvS[mlo];       Vp[o1] = v1a[j] + bvS[16 + mlo];
    }
}

// ------------------------------------------------- per-graph attention + FC head
// 192 threads = 6 waves; wave j owns query slot j, lane = feature d.
__global__ __launch_bounds__(192)
void attention(const float* __restrict__ x, const int* __restrict__ shuf,
               const float* __restrict__ keys, const float* __restrict__ Kp,
               const float* __restrict__ Vp,
               const float* __restrict__ wq, const float* __restrict__ bq,
               const float* __restrict__ wo, const float* __restrict__ bo,
               const float* __restrict__ wfc, const float* __restrict__ bfc,
               float* __restrict__ out) {
    __shared__ int   ordS[6];
    __shared__ float qs[6 * 32];
    __shared__ float sc[6 * 56];
    __shared__ float osm[6 * 32];
    __shared__ float flat[192];
    const int b = blockIdx.x;
    const int tid = threadIdx.x, j = tid >> 5, lane = tid & 31;
    const long base = (long)b * NNODES;

    if (tid == 0) {  // indices of the (exactly 6) mask==1 nodes, ascending
        int c = 0;
        for (int i = 0; i < NNODES; ++i)
            if (x[(base + i) * INCH + (INCH - 3)] > 0.5f) { if (c < 6) ordS[c] = i; ++c; }
        for (; c < 6; ++c) ordS[c] = 0;
    }
    __syncthreads();

    const int qrow = ordS[shuf[b * 6 + j]];
    qs[j * 32 + lane] = keys[(base + qrow) * HID + lane];
    __syncthreads();

    float Qd = bq[lane];
#pragma unroll
    for (int k = 0; k < HID; ++k) Qd += qs[j * 32 + k] * wq[k * HID + lane];

    const float scal = 0.17677669529663687f;  // 1/sqrt(32)
    for (int i = 0; i < NNODES; ++i) {
        float p = Qd * Kp[(base + i) * HID + lane];
        for (int off = 16; off; off >>= 1) p += __shfl_xor(p, off, 32);
        if (lane == 0) sc[j * 56 + i] = p * scal;
    }
    __syncthreads();

    float mx = -3.4e38f;
    for (int i = lane; i < NNODES; i += 32) mx = fmaxf(mx, sc[j * 56 + i]);
    for (int off = 16; off; off >>= 1) mx = fmaxf(mx, __shfl_xor(mx, off, 32));
    float sum = 0.f;
    for (int i = lane; i < NNODES; i += 32) {
        float e = __expf(sc[j * 56 + i] - mx);
        sc[j * 56 + i] = e;
        sum += e;
    }
    for (int off = 16; off; off >>= 1) sum += __shfl_xor(sum, off, 32);
    const float invs = 1.f / sum;
    __syncthreads();

    float od = 0.f;
    for (int i = 0; i < NNODES; ++i) od += sc[j * 56 + i] * Vp[(base + i) * HID + lane];
    osm[j * 32 + lane] = od * invs;
    __syncthreads();

    float o2 = bo[lane];
#pragma unroll
    for (int k = 0; k < HID; ++k) o2 += osm[j * 32 + k] * wo[k * HID + lane];
    flat[tid] = selu_f(o2);
    __syncthreads();

    if (tid < 6) {
        float acc = bfc[tid];
        for (int t = 0; t < 192; ++t) acc += flat[t] * wfc[t * 6 + tid];
        out[(long)b * 6 + tid] = acc;
    }
}

// ---------------------------------------------------------------- host
extern "C" void kernel_launch(void* const* d_in, const int* in_sizes, int n_in,
                              void* d_out, int out_size, void* d_ws, size_t ws_size,
                              hipStream_t stream) {
    const float* x    = (const float*)d_in[0];
    const int*   ei   = (const int*)  d_in[1];
    const int*   shuf = (const int*)  d_in[2];
    const float* wl0  = (const float*)d_in[3];
    const float* bl0  = (const float*)d_in[4];
    const float* wr0  = (const float*)d_in[5];
    const float* wlh  = (const float*)d_in[6];
    const float* blh  = (const float*)d_in[7];
    const float* wrh  = (const float*)d_in[8];
    const float* wq   = (const float*)d_in[9];
    const float* bq   = (const float*)d_in[10];
    const float* wk   = (const float*)d_in[11];
    const float* bk   = (const float*)d_in[12];
    const float* wv   = (const float*)d_in[13];
    const float* bv   = (const float*)d_in[14];
    const float* wo   = (const float*)d_in[15];
    const float* bo   = (const float*)d_in[16];
    const float* wfc  = (const float*)d_in[17];
    const float* bfc  = (const float*)d_in[18];
    float* out = (float*)d_out;

    float* ws  = (float*)d_ws;
    float* hA  = ws;
    float* hB  = hA + NTOT * 32;
    float* jk  = hB + NTOT * 32;
    float* agg = jk + NTOT * 32;
    float* cnt = agg + NTOT * 32;

    const int* src = ei;
    const int* dst = ei + EDGES;

    auto cdiv = [](long a, long b) -> unsigned int { return (unsigned int)((a + b - 1) / b); };

    // in-degrees (layer-invariant)
    zero_f4<<<cdiv(NTOT / 4, 256), 256, 0, stream>>>((float4*)cnt, NTOT / 4);
    count_deg<<<cdiv(EDGES, 256), 256, 0, stream>>>(dst, cnt, EDGES);

    // layer 0: K=13 (padded to 16 in agg)
    zero_f4<<<cdiv(NTOT * 16 / 4, 256), 256, 0, stream>>>((float4*)agg, NTOT * 16 / 4);
    scatter13<<<cdiv(EDGES * 16, 256), 256, 0, stream>>>(x, src, dst, agg, EDGES);
    sage_node<4, 13, 13, 16, true><<<(unsigned)(NTOT / 128), 256, 0, stream>>>(
        agg, cnt, x, wl0, bl0, wr0, hA, jk);

    // layers 1..4: K=32
    const float* prev = hA;
    float* cur = hB;
    for (int l = 0; l < 4; ++l) {
        zero_f4<<<cdiv(NTOT * 32 / 4, 256), 256, 0, stream>>>((float4*)agg, NTOT * 8);
        scatter32<<<cdiv(EDGES * 32, 256), 256, 0, stream>>>(prev, src, dst, agg, EDGES);
        sage_node<8, 32, 32, 32, false><<<(unsigned)(NTOT / 128), 256, 0, stream>>>(
            agg, cnt, prev, wlh + l * HID * HID, blh + l * HID, wrh + l * HID * HID, cur, jk);
        float* t = (float*)prev; prev = cur; cur = t;
    }

    // keys = selu(jk); K/V projections via WMMA; attention head
    selu_inplace<<<cdiv(NTOT * 32, 256), 256, 0, stream>>>(jk, NTOT * 32);
    proj_kv<<<(unsigned)(NTOT / 128), 256, 0, stream>>>(jk, wk, bk, wv, bv, hA, hB);
    attention<<<BGRAPH, 192, 0, stream>>>(x, shuf, jk, hA, hB, wq, bq, wo, bo, wfc, bfc, out);
}